// SyntheticDatasetModel_2688649527319
// MI455X (gfx1250) — compile-verified
//
#include <hip/hip_runtime.h>
#include <hip/hip_bf16.h>

typedef __attribute__((ext_vector_type(2))) float v2f;
typedef __attribute__((ext_vector_type(8))) float v8f;

#define N_NODES  50000
#define N_EDGES  800000
#define N_GRAPHS 500

// ---------------------------------------------------------------------------
// WMMA f32 GEMM: D[M,N] = A[M,K] @ B[K,N] (+bias per column) (optional relu)
// One wave (32 threads) per 16x16 output tile, K stepped by 4 via
// V_WMMA_F32_16X16X4_F32.  K % 4 == 0, N % 16 == 0 required; M is bounds-
// checked (clamped loads, guarded stores).
// A-matrix lane layout (ISA 7.12.2): lanes 0-15 carry K=k+0,k+1 for row M=l16,
// lanes 16-31 carry K=k+2,k+3 for the same rows.  B mirrors it; C/D: VGPR r,
// lanes 0-15 -> M=r, lanes 16-31 -> M=r+8, N = lane%16.
// ---------------------------------------------------------------------------
__global__ void __launch_bounds__(32)
gemm_wmma_f32(const float* __restrict__ A, const float* __restrict__ B,
              const float* __restrict__ bias, float* __restrict__ D,
              int M, int K, int N, int relu) {
  const int lane = threadIdx.x & 31;
  const int half = lane >> 4;   // 0: K pair {0,1}, 1: K pair {2,3}
  const int l16  = lane & 15;
  const int row0 = blockIdx.x << 4;
  const int col  = (blockIdx.y << 4) + l16;

  int arow = row0 + l16;
  if (arow >= M) arow = M - 1;                    // clamp (stores guarded)
  const float* Ap = A + (size_t)arow * K + (half << 1);
  const float* Bp = B + (size_t)(half << 1) * N + col;

  const float c0 = bias ? bias[col] : 0.0f;
  v8f acc;
#pragma unroll
  for (int r = 0; r < 8; ++r) acc[r] = c0;

  for (int k = 0; k < K; k += 4) {
    v2f a, b;
    a.x = Ap[k];
    a.y = Ap[k + 1];
    b.x = Bp[(size_t)k * N];
    b.y = Bp[(size_t)(k + 1) * N];
    acc = __builtin_amdgcn_wmma_f32_16x16x4_f32(false, a, false, b,
                                                (short)0, acc, false, false);
  }

#pragma unroll
  for (int r = 0; r < 8; ++r) {
    int srow = row0 + r + (half << 3);
    if (srow < M) {
      float v = acc[r];
      if (relu) v = fmaxf(v, 0.0f);
      D[(size_t)srow * N + col] = v;
    }
  }
}

// ---------------------------------------------------------------------------
// Degree / normalization
// ---------------------------------------------------------------------------
__global__ void deg_init_kernel(float* deg) {
  int v = blockIdx.x * blockDim.x + threadIdx.x;
  if (v < N_NODES) deg[v] = 1.0f;                 // self loop
}

__global__ void deg_accum_kernel(const long long* __restrict__ ei, float* deg) {
  int e = blockIdx.x * blockDim.x + threadIdx.x;
  if (e < N_EDGES) atomicAdd(&deg[(int)ei[(size_t)N_EDGES + e]], 1.0f);
}

__global__ void dinv_kernel(const float* __restrict__ deg, float* dinv) {
  int v = blockIdx.x * blockDim.x + threadIdx.x;
  if (v < N_NODES) dinv[v] = rsqrtf(fmaxf(deg[v], 1.0f));
}

// Precompute int32 endpoints + per-edge symmetric norm (read once per layer
// chunk-thread instead of re-reading int64 pairs + two dinv values).
__global__ void edge_prep_kernel(const long long* __restrict__ ei,
                                 const float* __restrict__ dinv,
                                 int* __restrict__ src32, int* __restrict__ dst32,
                                 float* __restrict__ enorm) {
  int e = blockIdx.x * blockDim.x + threadIdx.x;
  if (e >= N_EDGES) return;
  int s = (int)ei[e];
  int d = (int)ei[(size_t)N_EDGES + e];
  src32[e] = s;
  dst32[e] = d;
  enorm[e] = dinv[s] * dinv[d];
}

// out[v,c] = b[c] + hW[v,c] * dinv[v]^2   (bias + self-loop term)
__global__ void init_out_kernel(const float* __restrict__ hW,
                                const float* __restrict__ dinv,
                                const float* __restrict__ b,
                                float* __restrict__ out, int C) {
  size_t idx = (size_t)blockIdx.x * blockDim.x + threadIdx.x;
  size_t n = (size_t)N_NODES * C;
  if (idx >= n) return;
  int v = (int)(idx / C);
  int c = (int)(idx % C);
  float dv = dinv[v];
  out[idx] = b[c] + hW[idx] * dv * dv;
}

// out[dst] += hW[src] * norm[e]; each thread owns one edge x 4 channels
// (coalesced float4 gather, 4 float atomics).  chunks = C/4 divides 256.
__global__ void scatter_kernel(const float* __restrict__ hW,
                               const int* __restrict__ src32,
                               const int* __restrict__ dst32,
                               const float* __restrict__ enorm,
                               float* __restrict__ out, int C) {
  int chunks = C >> 2;
  int epb = 256 / chunks;                          // edges per block
  int e = blockIdx.x * epb + (int)threadIdx.x / chunks;
  if (e >= N_EDGES) return;
  int c4 = ((int)threadIdx.x % chunks) << 2;
  int s = src32[e], d = dst32[e];
  float w = enorm[e];
  const float4 val = *(const float4*)(hW + (size_t)s * C + c4);
  float* o = out + (size_t)d * C + c4;
  atomicAdd(o + 0, val.x * w);
  atomicAdd(o + 1, val.y * w);
  atomicAdd(o + 2, val.z * w);
  atomicAdd(o + 3, val.w * w);
}

__global__ void relu_kernel(float* h, size_t n) {
  size_t idx = (size_t)blockIdx.x * blockDim.x + threadIdx.x;
  if (idx < n) h[idx] = fmaxf(h[idx], 0.0f);
}

// ---------------------------------------------------------------------------
// Pooling + head
// ---------------------------------------------------------------------------
__global__ void zero_kernel(float* p, size_t n) {
  size_t idx = (size_t)blockIdx.x * blockDim.x + threadIdx.x;
  if (idx < n) p[idx] = 0.0f;
}

__global__ void pool_cnt_kernel(const long long* __restrict__ batch, float* cnt) {
  int v = blockIdx.x * blockDim.x + threadIdx.x;
  if (v < N_NODES) atomicAdd(&cnt[(int)batch[v]], 1.0f);
}

__global__ void pool_sum_kernel(const float* __restrict__ h,
                                const long long* __restrict__ batch,
                                float* pooled) {
  size_t idx = (size_t)blockIdx.x * blockDim.x + threadIdx.x;
  if (idx >= (size_t)N_NODES * 256) return;
  int v = (int)(idx >> 8);
  int c = (int)(idx & 255);
  atomicAdd(&pooled[(size_t)((int)batch[v]) * 256 + c], h[idx]);
}

__global__ void pool_div_kernel(float* pooled, const float* __restrict__ cnt) {
  int idx = blockIdx.x * blockDim.x + threadIdx.x;
  if (idx < N_GRAPHS * 256) pooled[idx] /= fmaxf(cnt[idx >> 8], 1.0f);
}

// out[g,t] = bm2[t] + sum_k z[g,k] * Wm2[k,t]   (500x10, K=256) — VALU is fine
__global__ void mlp2_kernel(const float* __restrict__ z,
                            const float* __restrict__ W,
                            const float* __restrict__ b, float* out) {
  int idx = blockIdx.x * blockDim.x + threadIdx.x;
  if (idx >= N_GRAPHS * 10) return;
  int g = idx / 10, t = idx % 10;
  float s = b[t];
  const float* zp = z + (size_t)g * 256;
  for (int k = 0; k < 256; ++k) s += zp[k] * W[k * 10 + t];
  out[idx] = s;
}

// ---------------------------------------------------------------------------
extern "C" void kernel_launch(void* const* d_in, const int* in_sizes, int n_in,
                              void* d_out, int out_size, void* d_ws, size_t ws_size,
                              hipStream_t stream) {
  const float*     x     = (const float*)d_in[0];
  const long long* ei    = (const long long*)d_in[1];
  const long long* batch = (const long long*)d_in[2];
  const float* Ws[4] = {(const float*)d_in[3], (const float*)d_in[5],
                        (const float*)d_in[7], (const float*)d_in[9]};
  const float* bs[4] = {(const float*)d_in[4], (const float*)d_in[6],
                        (const float*)d_in[8], (const float*)d_in[10]};
  const float* Wm1 = (const float*)d_in[11];
  const float* bm1 = (const float*)d_in[12];
  const float* Wm2 = (const float*)d_in[13];
  const float* bm2 = (const float*)d_in[14];
  float* out = (float*)d_out;

  // Workspace carve-up (all 16B-aligned regions).
  float* ws     = (float*)d_ws;
  float* deg    = ws;                              // N
  float* dinv   = deg  + N_NODES;                  // N
  float* buf0   = dinv + N_NODES;                  // N*256 (h ping-pong)
  float* bufW   = buf0 + (size_t)N_NODES * 256;    // N*256 (h @ W)
  float* pooled = bufW + (size_t)N_NODES * 256;    // G*256
  float* cnt    = pooled + (size_t)N_GRAPHS * 256; // G
  float* zbuf   = cnt + 512;                       // G*256 (pad cnt to 512)
  float* enorm  = zbuf + (size_t)N_GRAPHS * 256;   // E
  int*   src32  = (int*)(enorm + N_EDGES);         // E
  int*   dst32  = src32 + N_EDGES;                 // E

  const int dims[5] = {128, 32, 64, 128, 256};

  // Degree + symmetric norm
  deg_init_kernel<<<(N_NODES + 255) / 256, 256, 0, stream>>>(deg);
  deg_accum_kernel<<<(N_EDGES + 255) / 256, 256, 0, stream>>>(ei, deg);
  dinv_kernel<<<(N_NODES + 255) / 256, 256, 0, stream>>>(deg, dinv);
  edge_prep_kernel<<<(N_EDGES + 255) / 256, 256, 0, stream>>>(ei, dinv, src32, dst32, enorm);

  // 4 GCN layers; output of layer l overwrites buf0 (safe: gemm has already
  // consumed it into bufW by then — stream-ordered).
  const float* hin = x;
  for (int l = 0; l < 4; ++l) {
    int K = dims[l], C = dims[l + 1];
    gemm_wmma_f32<<<dim3(N_NODES / 16, C / 16), 32, 0, stream>>>(
        hin, Ws[l], nullptr, bufW, N_NODES, K, C, 0);
    size_t nelem = (size_t)N_NODES * C;
    init_out_kernel<<<(unsigned)((nelem + 255) / 256), 256, 0, stream>>>(
        bufW, dinv, bs[l], buf0, C);
    int epb = 256 / (C >> 2);
    scatter_kernel<<<(N_EDGES + epb - 1) / epb, 256, 0, stream>>>(
        bufW, src32, dst32, enorm, buf0, C);
    relu_kernel<<<(unsigned)((nelem + 255) / 256), 256, 0, stream>>>(buf0, nelem);
    hin = buf0;
  }

  // Global mean pool (pooled and cnt are contiguous -> single zero pass)
  zero_kernel<<<(N_GRAPHS * 256 + 512 + 255) / 256, 256, 0, stream>>>(
      pooled, (size_t)N_GRAPHS * 256 + 512);
  pool_cnt_kernel<<<(N_NODES + 255) / 256, 256, 0, stream>>>(batch, cnt);
  pool_sum_kernel<<<(unsigned)(((size_t)N_NODES * 256 + 255) / 256), 256, 0, stream>>>(
      buf0, batch, pooled);
  pool_div_kernel<<<(N_GRAPHS * 256 + 255) / 256, 256, 0, stream>>>(pooled, cnt);

  // MLP head: Linear(256,256)+ReLU via WMMA (bias folded into accumulator),
  // then tiny Linear(256,10).
  gemm_wmma_f32<<<dim3((N_GRAPHS + 15) / 16, 256 / 16), 32, 0, stream>>>(
      pooled, Wm1, bm1, zbuf, N_GRAPHS, 256, 256, 1);
  mlp2_kernel<<<(N_GRAPHS * 10 + 255) / 256, 256, 0, stream>>>(zbuf, Wm2, bm2, out);
}